// XhatModule_51728586113249
// MI455X (gfx1250) — compile-verified
//
#include <hip/hip_runtime.h>
#include <math.h>

typedef float v2f __attribute__((ext_vector_type(2)));
typedef float v8f __attribute__((ext_vector_type(8)));

#define NBLK 128
#define NTHR 256
#define NWAVE (NTHR / 32)
#define GN_MAX_ITER 300
#define GN_MAX_LS 25
#define GN_TOL 1e-9f

struct GNState {
    unsigned cnt, gen, pad0, pad1;
    int done, lsDone, pad2, pad3;
    float mu, loss, gd, step;
    float u[4], du[4], x[4], sd[4], xn[4];
};

// ---------- scalar helpers (accurate path, used on the 4-parameter state) ----
__device__ __forceinline__ float sp_acc(float u)   { return log1pf(expf(u)); }
__device__ __forceinline__ float sigm_acc(float u) { return 1.0f / (1.0f + expf(-u)); }

// ---------- device-wide sense-reversing barrier ------------------------------
__device__ __forceinline__ void grid_sync(GNState* S) {
    __threadfence();
    __syncthreads();
    if (threadIdx.x == 0) {
        volatile unsigned* gen = &S->gen;
        unsigned g = *gen;
        if (atomicAdd(&S->cnt, 1u) == (unsigned)(NBLK - 1)) {
            S->cnt = 0u;               // safe: everyone has arrived, nobody re-arrives until gen bumps
            __threadfence();
            atomicAdd(&S->gen, 1u);    // release
        } else {
            while (*gen == g) { __builtin_amdgcn_s_sleep(2); }
        }
    }
    __threadfence();
    __syncthreads();
}

// ---------- WMMA column-sum: D[m][n] = sum_k B[k][n]  (A = all ones) ---------
__device__ __forceinline__ v8f wmma_colsum_acc(v8f acc, float b0, float b1) {
    v2f ones; ones.x = 1.0f; ones.y = 1.0f;
    v2f b;    b.x = b0;      b.y = b1;
    return __builtin_amdgcn_wmma_f32_16x16x4_f32(
        /*neg_a=*/false, ones, /*neg_b=*/false, b,
        /*c_mod=*/(short)0, acc, /*reuse_a=*/false, /*reuse_b=*/false);
}

// ---------- init kernel ------------------------------------------------------
__global__ void gn_init(const float* __restrict__ log_mu,
                        const float* __restrict__ x_init,
                        GNState* S) {
    if (threadIdx.x == 0 && blockIdx.x == 0) {
        S->cnt = 0u; S->gen = 0u;
        S->done = 0;  S->lsDone = 0;
        float mu = expf(log_mu[0]);
        S->mu = mu; S->step = 1.0f; S->loss = 0.0f; S->gd = 0.0f;
        #pragma unroll
        for (int j = 0; j < 4; ++j) {
            float xi = x_init[j];
            xi = fmaxf(xi, 1e-3f);                 // clip in gn_solver
            xi = fmaxf(xi, 1e-6f);                 // clip inside inv_sp
            float uj = logf(expf(xi) - 1.0f + 1e-8f);  // inv_softplus
            S->u[j]  = uj;
            S->du[j] = 0.0f;
            S->x[j]  = sp_acc(uj);
            S->sd[j] = sigm_acc(uj);
            S->xn[j] = S->x[j];
        }
    }
}

// ---------- persistent Gauss-Newton kernel -----------------------------------
__global__ void __launch_bounds__(NTHR)
gn_persistent(const float* __restrict__ y,
              const float* __restrict__ t,
              float* __restrict__ out,
              GNState* S,
              float* __restrict__ qpart,   // [NBLK*16] per-block Gram partials
              float* __restrict__ lpart,   // [NBLK]    per-block line-search partials
              int N) {
    volatile GNState* vS = (volatile GNState*)S;
    __shared__ float red[NWAVE * 16];
    __shared__ float sq[16];
    __shared__ float sls[NWAVE];

    const int lane = threadIdx.x & 31;
    const int wid  = threadIdx.x >> 5;
    const int col  = lane & 15;
    const int half = lane >> 4;          // 0 or 1
    const int gtid = blockIdx.x * NTHR + threadIdx.x;
    const int gstride = NBLK * NTHR;
    const bool isB0T0 = (blockIdx.x == 0 && threadIdx.x == 0);

    for (int it = 0; it < GN_MAX_ITER; ++it) {
        grid_sync(S);
        if (vS->done) break;

        // ---------------- pass A: Gram / gradient / loss reductions ----------
        const float c0 = vS->x[0], c1 = vS->x[1], l0 = vS->x[2], l1 = vS->x[3];
        const float s0 = vS->sd[0], s1 = vS->sd[1], s2 = vS->sd[2], s3 = vS->sd[3];

        float q[15];
        #pragma unroll
        for (int i = 0; i < 15; ++i) q[i] = 0.0f;

        for (int i = gtid; i < N; i += gstride) {
            __builtin_prefetch(t + i + gstride, 0, 1);   // global_prefetch_b8
            __builtin_prefetch(y + i + gstride, 0, 1);
            float ti = t[i], yi = y[i];
            float e0 = __expf(-l0 * ti);
            float e1 = __expf(-l1 * ti);
            float m0 = c0 * e0, m1 = c1 * e1;
            float r  = yi - m0 - m1;
            float j0 = e0 * s0;
            float j1 = e1 * s1;
            float j2 = -ti * m0 * s2;
            float j3 = -ti * m1 * s3;
            q[0]  = fmaf(j0, j0, q[0]);
            q[1]  = fmaf(j0, j1, q[1]);
            q[2]  = fmaf(j0, j2, q[2]);
            q[3]  = fmaf(j0, j3, q[3]);
            q[4]  = fmaf(j1, j1, q[4]);
            q[5]  = fmaf(j1, j2, q[5]);
            q[6]  = fmaf(j1, j3, q[6]);
            q[7]  = fmaf(j2, j2, q[7]);
            q[8]  = fmaf(j2, j3, q[8]);
            q[9]  = fmaf(j3, j3, q[9]);
            q[10] = fmaf(j0, r,  q[10]);
            q[11] = fmaf(j1, r,  q[11]);
            q[12] = fmaf(j2, r,  q[12]);
            q[13] = fmaf(j3, r,  q[13]);
            q[14] = fmaf(r,  r,  q[14]);
        }
        // intra-wave (wave32) butterfly reduce
        #pragma unroll
        for (int i = 0; i < 15; ++i) {
            q[i] += __shfl_xor(q[i], 16);
            q[i] += __shfl_xor(q[i], 8);
            q[i] += __shfl_xor(q[i], 4);
            q[i] += __shfl_xor(q[i], 2);
            q[i] += __shfl_xor(q[i], 1);
        }
        if (lane == 0) {
            #pragma unroll
            for (int i = 0; i < 15; ++i) red[wid * 16 + i] = q[i];
            red[wid * 16 + 15] = 0.0f;
        }
        __syncthreads();
        // cross-wave reduce (8 wave-partials of 16) via two WMMAs, wave 0 only
        if (wid == 0) {
            v8f d = {};
            d = wmma_colsum_acc(d, red[(half * 2 + 0) * 16 + col],
                                   red[(half * 2 + 1) * 16 + col]);
            d = wmma_colsum_acc(d, red[(4 + half * 2 + 0) * 16 + col],
                                   red[(4 + half * 2 + 1) * 16 + col]);
            if (lane < 16) qpart[blockIdx.x * 16 + lane] = d[0];  // row M=0 of D
        }
        grid_sync(S);

        // ---------------- block 0: cross-block WMMA reduce + 4x4 solve -------
        if (blockIdx.x == 0) {
            if (wid == 0) {
                v8f d = {};
                for (int g = 0; g < NBLK; g += 4) {
                    d = wmma_colsum_acc(d, qpart[(g + half * 2 + 0) * 16 + col],
                                           qpart[(g + half * 2 + 1) * 16 + col]);
                }
                if (lane < 16) sq[lane] = d[0];
            }
            __syncthreads();
            if (threadIdx.x == 0) {
                float mu  = S->mu;
                float reg = mu + 1e-7f;
                float x0 = S->x[0], x1 = S->x[1], x2 = S->x[2], x3 = S->x[3];
                float gv[4];
                gv[0] = -sq[10] + mu * x0 * S->sd[0];
                gv[1] = -sq[11] + mu * x1 * S->sd[1];
                gv[2] = -sq[12] + mu * x2 * S->sd[2];
                gv[3] = -sq[13] + mu * x3 * S->sd[3];
                float A[4][5];
                A[0][0] = sq[0] + reg; A[0][1] = sq[1];       A[0][2] = sq[2];       A[0][3] = sq[3];
                A[1][0] = sq[1];       A[1][1] = sq[4] + reg; A[1][2] = sq[5];       A[1][3] = sq[6];
                A[2][0] = sq[2];       A[2][1] = sq[5];       A[2][2] = sq[7] + reg; A[2][3] = sq[8];
                A[3][0] = sq[3];       A[3][1] = sq[6];       A[3][2] = sq[8];       A[3][3] = sq[9] + reg;
                #pragma unroll
                for (int i = 0; i < 4; ++i) A[i][4] = -gv[i];
                // Gaussian elimination with partial pivoting
                for (int k = 0; k < 4; ++k) {
                    int p = k; float mx = fabsf(A[k][k]);
                    for (int r = k + 1; r < 4; ++r) {
                        float v = fabsf(A[r][k]);
                        if (v > mx) { mx = v; p = r; }
                    }
                    if (p != k) {
                        for (int j = k; j < 5; ++j) {
                            float tmp = A[k][j]; A[k][j] = A[p][j]; A[p][j] = tmp;
                        }
                    }
                    float inv = 1.0f / A[k][k];
                    for (int r = k + 1; r < 4; ++r) {
                        float f = A[r][k] * inv;
                        for (int j = k; j < 5; ++j) A[r][j] -= f * A[k][j];
                    }
                }
                float du[4];
                for (int k = 3; k >= 0; --k) {
                    float s = A[k][4];
                    for (int j = k + 1; j < 4; ++j) s -= A[k][j] * du[j];
                    du[k] = s / A[k][k];
                }
                float gd = gv[0]*du[0] + gv[1]*du[1] + gv[2]*du[2] + gv[3]*du[3];
                float loss = sq[14] + mu * (x0*x0 + x1*x1 + x2*x2 + x3*x3);
                #pragma unroll
                for (int j = 0; j < 4; ++j) {
                    S->du[j] = du[j];
                    S->xn[j] = sp_acc(S->u[j] + du[j]);   // first trial, step = 1
                }
                S->gd = gd; S->loss = loss; S->step = 1.0f; S->lsDone = 0;
            }
        }
        grid_sync(S);

        // ---------------- backtracking Armijo line search --------------------
        for (int k = 0; k < GN_MAX_LS; ++k) {
            const float a0 = vS->xn[0], a1 = vS->xn[1];
            const float b0 = vS->xn[2], b1 = vS->xn[3];
            float ls = 0.0f;
            for (int i = gtid; i < N; i += gstride) {
                float ti = t[i], yi = y[i];
                float rn = yi - a0 * __expf(-b0 * ti) - a1 * __expf(-b1 * ti);
                ls = fmaf(rn, rn, ls);
            }
            ls += __shfl_xor(ls, 16);
            ls += __shfl_xor(ls, 8);
            ls += __shfl_xor(ls, 4);
            ls += __shfl_xor(ls, 2);
            ls += __shfl_xor(ls, 1);
            if (lane == 0) sls[wid] = ls;
            __syncthreads();
            if (threadIdx.x == 0) {
                float s = 0.0f;
                #pragma unroll
                for (int w = 0; w < NWAVE; ++w) s += sls[w];
                lpart[blockIdx.x] = s;
            }
            __syncthreads();
            grid_sync(S);
            if (isB0T0) {
                float s = 0.0f;
                for (int b = 0; b < NBLK; ++b) s += lpart[b];     // fixed order: deterministic
                float n0 = S->xn[0], n1 = S->xn[1], n2 = S->xn[2], n3 = S->xn[3];
                float loss_new = s + S->mu * (n0*n0 + n1*n1 + n2*n2 + n3*n3);
                float step = S->step;
                bool acc = (loss_new <= S->loss - 1e-4f * step * S->gd);
                if (acc) {
                    S->lsDone = 1;
                } else {
                    step *= 0.5f;                // reference halves on every rejection
                    S->step = step;
                    if (k == GN_MAX_LS - 1) {
                        S->lsDone = 1;           // 25 failures -> step = 2^-25
                    } else {
                        #pragma unroll
                        for (int j = 0; j < 4; ++j)
                            S->xn[j] = sp_acc(S->u[j] + step * S->du[j]);
                    }
                }
            }
            grid_sync(S);
            if (vS->lsDone) break;
        }

        // ---------------- accept step, convergence check ---------------------
        if (isB0T0) {
            float step = S->step;
            float nd2 = 0.0f;
            #pragma unroll
            for (int j = 0; j < 4; ++j) {
                float dj = S->du[j];
                nd2 += dj * dj;
                S->u[j] += step * dj;
            }
            if (step * sqrtf(nd2) < GN_TOL) S->done = 1;
            #pragma unroll
            for (int j = 0; j < 4; ++j) {
                float uj = S->u[j];
                S->x[j]  = sp_acc(uj);
                S->sd[j] = sigm_acc(uj);
            }
        }
    }

    if (isB0T0) {
        #pragma unroll
        for (int j = 0; j < 4; ++j) out[j] = sp_acc(S->u[j]);
    }
}

// ---------- entry point ------------------------------------------------------
extern "C" void kernel_launch(void* const* d_in, const int* in_sizes, int n_in,
                              void* d_out, int out_size, void* d_ws, size_t ws_size,
                              hipStream_t stream) {
    const float* log_mu = (const float*)d_in[0];
    const float* y      = (const float*)d_in[1];
    const float* t      = (const float*)d_in[2];
    const float* x_init = (const float*)d_in[3];
    float* out = (float*)d_out;
    int N = in_sizes[1];

    char* ws = (char*)d_ws;
    GNState* S    = (GNState*)ws;
    float* qpart  = (float*)(ws + 512);
    float* lpart  = qpart + NBLK * 16;

    gn_init<<<1, 1, 0, stream>>>(log_mu, x_init, S);
    gn_persistent<<<NBLK, NTHR, 0, stream>>>(y, t, out, S, qpart, lpart, N);
}